// GNN_32933809226560
// MI455X (gfx1250) — compile-verified
//
#include <hip/hip_runtime.h>
#include <hip/hip_bf16.h>
#include <math.h>

// Problem constants (match reference)
#define NN 50000
#define EE 800000
#define FF 128
#define DH 128
#define RR 6
#define EPS 1e-5f
#define NEG_SLOPE 0.01f

typedef __attribute__((ext_vector_type(2))) float v2f;
typedef __attribute__((ext_vector_type(8))) float v8f;

// ---------------------------------------------------------------------------
// Generic f32 WMMA GEMM: C[M,128] = A[M,128] @ B[128,128] (+ bias)
// One wave computes a 16x16 tile of C via 32 chained V_WMMA_F32_16X16X4_F32.
// VGPR layouts per CDNA5 ISA 7.12.2:
//   A 16x4 f32: lanes 0-15 hold M=0..15 {K=k0,k0+1}; lanes 16-31 hold {K=k0+2,k0+3}
//   B 4x16 f32: VGPR0 lanes 0-15 = row k0, lanes 16-31 = row k0+2; VGPR1 = k0+1 / k0+3
//   C 16x16 f32: VGPR p, lanes 0-15 -> (M=p, N=lane); lanes 16-31 -> (M=p+8, N=lane-16)
// ---------------------------------------------------------------------------
__global__ __launch_bounds__(256)
void gemm128_wmma(const float* __restrict__ A, const float* __restrict__ B,
                  const float* __restrict__ bias, float* __restrict__ C, int M)
{
    const int K = 128, NO = 128;
    int wave = threadIdx.x >> 5;   // 0..7 -> column tile
    int lane = threadIdx.x & 31;
    int half = lane >> 4;          // 0 or 1
    int l    = lane & 15;
    int row0 = blockIdx.x * 16;
    int col0 = wave * 16;
    if (row0 >= M) return;         // uniform across block; EXEC stays all-ones

    const float* __restrict__ arow = A + (size_t)(row0 + l) * K;

    v8f acc = {};
    #pragma unroll 4
    for (int k0 = 0; k0 < K; k0 += 4) {
        int kb = k0 + half * 2;
        v2f a, b;
        a.x = arow[kb];
        a.y = arow[kb + 1];
        b.x = B[(size_t)kb * NO + col0 + l];
        b.y = B[(size_t)(kb + 1) * NO + col0 + l];
        acc = __builtin_amdgcn_wmma_f32_16x16x4_f32(
            /*neg_a=*/false, a, /*neg_b=*/false, b,
            /*c_mod=*/(short)0, acc, /*reuse_a=*/false, /*reuse_b=*/false);
    }

    float bval = bias ? bias[col0 + l] : 0.0f;
    #pragma unroll
    for (int p = 0; p < 8; ++p) {
        int m = row0 + p + half * 8;
        C[(size_t)m * NO + col0 + l] = acc[p] + bval;
    }
}

// ---------------------------------------------------------------------------
// Fills
// ---------------------------------------------------------------------------
__global__ void fill_f32(float* __restrict__ p, float val, size_t n)
{
    size_t i = (size_t)blockIdx.x * blockDim.x + threadIdx.x;
    if (i < n) p[i] = val;
}

// ---------------------------------------------------------------------------
// RGCN: per-(dst, relation) incoming-edge counts
// ---------------------------------------------------------------------------
__global__ void edge_count(const int* __restrict__ dst, const int* __restrict__ et,
                           float* __restrict__ cnt, int E)
{
    int e = blockIdx.x * blockDim.x + threadIdx.x;
    if (e < E) atomicAdd(&cnt[(size_t)dst[e] * RR + et[e]], 1.0f);
}

// RGCN aggregation: h[dst] += xw_r[src] / max(cnt[dst,r],1).  One 128-thread
// block per edge; lane j owns feature j.
__global__ __launch_bounds__(128)
void rgcn_agg(const int* __restrict__ src, const int* __restrict__ dst,
              const int* __restrict__ et, const float* __restrict__ xw,
              const float* __restrict__ cnt, float* __restrict__ h, int E)
{
    int e = blockIdx.x;
    if (e >= E) return;
    int j = threadIdx.x;
    int s = src[e], d = dst[e], r = et[e];
    float c = cnt[(size_t)d * RR + r];
    float scale = 1.0f / fmaxf(c, 1.0f);
    float m = xw[((size_t)r * NN + s) * DH + j];
    atomicAdd(&h[(size_t)d * DH + j], m * scale);
}

// ---------------------------------------------------------------------------
// Edge attention score + segment max.  One wave32 per edge, lane = float4 of DH.
// ---------------------------------------------------------------------------
__device__ inline void atomicMaxFloat(float* addr, float value)
{
    if (value >= 0.0f)
        atomicMax((int*)addr, __float_as_int(value));
    else
        atomicMin((unsigned int*)addr, __float_as_uint(value));
}

__global__ __launch_bounds__(256)
void edge_score(const int* __restrict__ src, const int* __restrict__ dst,
                const float* __restrict__ q, const float* __restrict__ k,
                float* __restrict__ score, float* __restrict__ smax, int E)
{
    int e = blockIdx.x * 8 + (threadIdx.x >> 5);
    int lane = threadIdx.x & 31;
    if (e >= E) return;
    int s = src[e], d = dst[e];
    const float4* qv = (const float4*)(q + (size_t)d * DH);
    const float4* kv = (const float4*)(k + (size_t)s * DH);
    float4 a = qv[lane];
    float4 b = kv[lane];
    float p = a.x * b.x + a.y * b.y + a.z * b.z + a.w * b.w;
    #pragma unroll
    for (int off = 16; off; off >>= 1) p += __shfl_xor(p, off, 32);
    if (lane == 0) {
        p *= 0.08838834764831845f;   // 1/sqrt(128)
        score[e] = p;
        atomicMaxFloat(&smax[d], p);
    }
}

// exp(score - smax[dst]) and segment-sum denominator (score overwritten in place)
__global__ void edge_expsum(const int* __restrict__ dst, float* __restrict__ score,
                            const float* __restrict__ smax, float* __restrict__ denom, int E)
{
    int e = blockIdx.x * blockDim.x + threadIdx.x;
    if (e < E) {
        int d = dst[e];
        float a = __expf(score[e] - smax[d]);
        score[e] = a;
        atomicAdd(&denom[d], a);
    }
}

// agg[dst] += (a/denom[dst]) * v[src].  One 128-thread block per edge.
__global__ __launch_bounds__(128)
void attn_agg(const int* __restrict__ src, const int* __restrict__ dst,
              const float* __restrict__ score, const float* __restrict__ denom,
              const float* __restrict__ v, float* __restrict__ agg, int E)
{
    int e = blockIdx.x;
    if (e >= E) return;
    int j = threadIdx.x;
    int s = src[e], d = dst[e];
    float alpha = score[e] / denom[d];
    atomicAdd(&agg[(size_t)d * DH + j], alpha * v[(size_t)s * DH + j]);
}

// ---------------------------------------------------------------------------
// BatchNorm: column sums / sum-of-squares (LDS accumulate, then global atomics)
// ---------------------------------------------------------------------------
__global__ __launch_bounds__(256)
void bn_stats(const float* __restrict__ h, float* __restrict__ sums,
              float* __restrict__ sqs, size_t n)
{
    __shared__ float ls[DH];
    __shared__ float lq[DH];
    if (threadIdx.x < DH) { ls[threadIdx.x] = 0.0f; lq[threadIdx.x] = 0.0f; }
    __syncthreads();
    for (size_t i = (size_t)blockIdx.x * blockDim.x + threadIdx.x; i < n;
         i += (size_t)gridDim.x * blockDim.x) {
        float v = h[i];
        int j = (int)(i & (DH - 1));
        atomicAdd(&ls[j], v);
        atomicAdd(&lq[j], v * v);
    }
    __syncthreads();
    if (threadIdx.x < DH) {
        atomicAdd(&sums[threadIdx.x], ls[threadIdx.x]);
        atomicAdd(&sqs[threadIdx.x], lq[threadIdx.x]);
    }
}

__global__ void bn_final(const float* __restrict__ h, const float* __restrict__ sums,
                         const float* __restrict__ sqs, const float* __restrict__ gamma,
                         const float* __restrict__ beta, float* __restrict__ out, size_t n)
{
    size_t i = (size_t)blockIdx.x * blockDim.x + threadIdx.x;
    if (i >= n) return;
    int j = (int)(i & (DH - 1));
    const float invN = 1.0f / (float)NN;
    float mu  = sums[j] * invN;
    float var = sqs[j] * invN - mu * mu;
    float x = (h[i] - mu) * rsqrtf(var + EPS);
    float y = gamma[j] * x + beta[j];
    out[i] = y > 0.0f ? y : NEG_SLOPE * y;
}

// ---------------------------------------------------------------------------
// Host driver
// ---------------------------------------------------------------------------
extern "C" void kernel_launch(void* const* d_in, const int* in_sizes, int n_in,
                              void* d_out, int out_size, void* d_ws, size_t ws_size,
                              hipStream_t stream)
{
    (void)in_sizes; (void)n_in; (void)out_size; (void)ws_size;

    const float* x        = (const float*)d_in[0];
    const int*   ei       = (const int*)d_in[1];      // [2,E]: row0=src, row1=dst
    const int*   et       = (const int*)d_in[2];
    const float* rgcn_w   = (const float*)d_in[3];    // [R,128,128]
    const float* rgcn_root= (const float*)d_in[4];
    const float* rgcn_b   = (const float*)d_in[5];
    const float* wq = (const float*)d_in[6];  const float* bq = (const float*)d_in[7];
    const float* wk = (const float*)d_in[8];  const float* bk = (const float*)d_in[9];
    const float* wv = (const float*)d_in[10]; const float* bv = (const float*)d_in[11];
    const float* ws_ = (const float*)d_in[12]; const float* bs = (const float*)d_in[13];
    const float* gamma = (const float*)d_in[14];
    const float* beta  = (const float*)d_in[15];
    float* out = (float*)d_out;

    const int* srcp = ei;
    const int* dstp = ei + EE;

    // Workspace carve-up (floats)
    float* w = (float*)d_ws;
    size_t off = 0;
    float* h     = w + off; off += (size_t)NN * DH;          // RGCN output
    float* xw    = w + off; off += (size_t)RR * NN * DH;     // x @ w_r, all relations
    float* cnt   = w + off; off += (size_t)NN * RR;
    float* q     = w + off; off += (size_t)NN * DH;
    float* kbuf  = w + off; off += (size_t)NN * DH;
    float* vbuf  = w + off; off += (size_t)NN * DH;
    float* agg   = w + off; off += (size_t)NN * DH;          // skip + attention agg
    float* score = w + off; off += (size_t)EE;
    float* smax  = w + off; off += (size_t)NN;
    float* denom = w + off; off += (size_t)NN;
    float* sums  = w + off; off += (size_t)DH;
    float* sqs   = w + off; off += (size_t)DH;

    const int TB = 256;
    const int gemm_blocks = NN / 16;   // 3125

    // Zero / init scratch
    fill_f32<<<((size_t)NN * RR + TB - 1) / TB, TB, 0, stream>>>(cnt, 0.0f, (size_t)NN * RR);
    fill_f32<<<(NN + TB - 1) / TB, TB, 0, stream>>>(denom, 0.0f, (size_t)NN);
    fill_f32<<<(NN + TB - 1) / TB, TB, 0, stream>>>(smax, -INFINITY, (size_t)NN);
    fill_f32<<<1, TB, 0, stream>>>(sums, 0.0f, (size_t)DH);
    fill_f32<<<1, TB, 0, stream>>>(sqs, 0.0f, (size_t)DH);

    // ---- RGCN ----
    gemm128_wmma<<<gemm_blocks, TB, 0, stream>>>(x, rgcn_root, rgcn_b, h, NN);
    for (int r = 0; r < RR; ++r)
        gemm128_wmma<<<gemm_blocks, TB, 0, stream>>>(
            x, rgcn_w + (size_t)r * FF * DH, nullptr, xw + (size_t)r * NN * DH, NN);

    edge_count<<<(EE + TB - 1) / TB, TB, 0, stream>>>(dstp, et, cnt, EE);
    rgcn_agg<<<EE, 128, 0, stream>>>(srcp, dstp, et, xw, cnt, h, EE);

    // ---- TransformerConv ----
    gemm128_wmma<<<gemm_blocks, TB, 0, stream>>>(h, wq,  bq, q,    NN);
    gemm128_wmma<<<gemm_blocks, TB, 0, stream>>>(h, wk,  bk, kbuf, NN);
    gemm128_wmma<<<gemm_blocks, TB, 0, stream>>>(h, wv,  bv, vbuf, NN);
    gemm128_wmma<<<gemm_blocks, TB, 0, stream>>>(h, ws_, bs, agg,  NN);  // skip proj seeds agg

    edge_score<<<EE / 8, TB, 0, stream>>>(srcp, dstp, q, kbuf, score, smax, EE);
    edge_expsum<<<(EE + TB - 1) / TB, TB, 0, stream>>>(dstp, score, smax, denom, EE);
    attn_agg<<<EE, 128, 0, stream>>>(srcp, dstp, score, denom, vbuf, agg, EE);

    // ---- BatchNorm + LeakyReLU ----
    bn_stats<<<512, TB, 0, stream>>>(agg, sums, sqs, (size_t)NN * DH);
    bn_final<<<((size_t)NN * DH + TB - 1) / TB, TB, 0, stream>>>(
        agg, sums, sqs, gamma, beta, out, (size_t)NN * DH);
}